// LiftSplatShoot_Depth_49855980372229
// MI455X (gfx1250) — compile-verified
//
#include <hip/hip_runtime.h>
#include <hip/hip_bf16.h>

// ---- problem constants (match reference) ----
#define NCAM 6
#define DBINS 41
#define CCH 64
#define FHh 56
#define FWw 100
#define NXV 33
#define NYV 33
#define NZV 2
#define PIXPC (FHh * FWw)          // 5600 pixels per camera
#define NPIX (NCAM * PIXPC)        // 33600 pixels total
#define CHSTRIDE (FHh * FWw)       // 5600 floats between channels of x
#define PLANE (NXV * NYV)          // 1089 (BEV spatial plane)

typedef __attribute__((ext_vector_type(2))) float v2f;
typedef __attribute__((ext_vector_type(8))) float v8f;

// ---------------- 3x3 inverse (adjugate) ----------------
__device__ __forceinline__ void inv3(const float* m, float* o) {
    float a = m[0], b = m[1], c = m[2];
    float d = m[3], e = m[4], f = m[5];
    float g = m[6], h = m[7], i = m[8];
    float A  = e * i - f * h;
    float Bm = -(d * i - f * g);
    float Cc = d * h - e * g;
    float det = a * A + b * Bm + c * Cc;
    float id = 1.0f / det;
    o[0] = A * id;             o[1] = -(b * i - c * h) * id;  o[2] = (b * f - c * e) * id;
    o[3] = Bm * id;            o[4] = (a * i - c * g) * id;   o[5] = -(a * f - c * d) * id;
    o[6] = Cc * id;            o[7] = -(a * h - b * g) * id;  o[8] = (a * e - b * d) * id;
}

// Per-camera precompute: cams[n*24 + ...] = [inv(post_rots) 9][post_trans 3][combine 9][trans 3]
__global__ void lss_prep_kernel(const float* __restrict__ rots,
                                const float* __restrict__ trans,
                                const float* __restrict__ intrins,
                                const float* __restrict__ post_rots,
                                const float* __restrict__ post_trans,
                                float* __restrict__ cams) {
    int n = threadIdx.x;
    if (n >= NCAM) return;
    float R[9], K[9], P[9], Pi[9], Ki[9], Cm[9];
    for (int i = 0; i < 9; ++i) {
        R[i] = rots[n * 9 + i];
        K[i] = intrins[n * 9 + i];
        P[i] = post_rots[n * 9 + i];
    }
    inv3(P, Pi);
    inv3(K, Ki);
    for (int r = 0; r < 3; ++r)
        for (int c = 0; c < 3; ++c) {
            float s = 0.0f;
            for (int k = 0; k < 3; ++k) s += R[r * 3 + k] * Ki[k * 3 + c];
            Cm[r * 3 + c] = s;
        }
    float* o = cams + n * 24;
    for (int i = 0; i < 9; ++i) o[i]      = Pi[i];
    for (int i = 0; i < 3; ++i) o[9 + i]  = post_trans[n * 3 + i];
    for (int i = 0; i < 9; ++i) o[12 + i] = Cm[i];
    for (int i = 0; i < 3; ++i) o[21 + i] = trans[n * 3 + i];
}

__global__ void lss_zero_kernel(float* __restrict__ out, int n) {
    int i = blockIdx.x * blockDim.x + threadIdx.x;
    if (i < n) out[i] = 0.0f;
}

// One wave (32 lanes) per pixel. 8 waves / block.
__global__ __launch_bounds__(256) void lss_splat_kernel(const float* __restrict__ x,
                                                        const float* __restrict__ cams,
                                                        float* __restrict__ out) {
    const int lane = threadIdx.x & 31;
    const int wv   = threadIdx.x >> 5;
    const int wid  = blockIdx.x * 8 + wv;          // pixel id, < NPIX exactly

    const int n   = wid / PIXPC;
    const int rem = wid - n * PIXPC;
    const int h   = rem / FWw;
    const int w   = rem - h * FWw;
    const int pix = h * FWw + w;

    __shared__ float    wbuf[8][44];
    __shared__ unsigned sbuf[8][44];

    // ---- per-camera constants ----
    const float* cam = cams + n * 24;
    float PR[9], PT[3], CM[9], TR[3];
#pragma unroll
    for (int i = 0; i < 9; ++i) PR[i] = cam[i];
#pragma unroll
    for (int i = 0; i < 3; ++i) PT[i] = cam[9 + i];
#pragma unroll
    for (int i = 0; i < 9; ++i) CM[i] = cam[12 + i];
#pragma unroll
    for (int i = 0; i < 3; ++i) TR[i] = cam[21 + i];

    const float* xcam = x + (size_t)n * (DBINS + CCH) * CHSTRIDE;

    // prefetch the ctx block for this pixel (gfx1250 global_prefetch_b8)
    __builtin_prefetch(xcam + (DBINS + lane) * CHSTRIDE + pix, 0, 1);
    __builtin_prefetch(xcam + (DBINS + 32 + lane) * CHSTRIDE + pix, 0, 1);

    // ---- softmax over 41 depth logits (lanes parallel over d) ----
    float l1 = xcam[lane * CHSTRIDE + pix];
    float l2 = (lane < 9) ? xcam[(32 + lane) * CHSTRIDE + pix] : -3.4e38f;
    float mx = fmaxf(l1, l2);
#pragma unroll
    for (int m = 16; m >= 1; m >>= 1) mx = fmaxf(mx, __shfl_xor(mx, m, 32));
    float e1 = __expf(l1 - mx);
    float e2 = (lane < 9) ? __expf(l2 - mx) : 0.0f;
    float sm = e1 + e2;
#pragma unroll
    for (int m = 16; m >= 1; m >>= 1) sm += __shfl_xor(sm, m, 32);
    float inv = 1.0f / sm;
    wbuf[wv][lane] = e1 * inv;
    if (lane < 9) wbuf[wv][32 + lane] = e2 * inv;

    // ---- ctx features: 2 channels per lane ----
    float ctx0 = xcam[(DBINS + lane) * CHSTRIDE + pix];
    float ctx1 = xcam[(DBINS + 32 + lane) * CHSTRIDE + pix];

    // ---- geometry: p1(d) = PRinv*(f - post_trans), f = (xs, ys, 4+d) ----
    const float xs = (float)w * (1599.0f / 99.0f);
    const float ys = (float)h * (899.0f / 55.0f);
    const float fx = xs - PT[0], fy = ys - PT[1], fz0 = -PT[2];
    const float bx = PR[0] * fx + PR[1] * fy + PR[2] * fz0;   // + PR[2]*ds
    const float by = PR[3] * fx + PR[4] * fy + PR[5] * fz0;   // + PR[5]*ds
    const float bz = PR[6] * fx + PR[7] * fy + PR[8] * fz0;   // + PR[8]*ds

    // ---- WMMA A operand: rows 0..2 = [combine | trans], padded to 16x4 ----
    // ISA layout: lanes 0-15 hold A[M=lane][K=0],A[M=lane][K=1];
    //             lanes 16-31 hold A[M=lane-16][K=2],A[M=lane-16][K=3]
    v2f aop = {0.0f, 0.0f};
    if (lane < 3) {
        aop[0] = CM[lane * 3 + 0];
        aop[1] = CM[lane * 3 + 1];
    } else if (lane >= 16 && lane < 19) {
        int r = lane - 16;
        aop[0] = CM[r * 3 + 2];
        aop[1] = TR[r];
    }

    // ---- 3x WMMA: 16 depth bins per v_wmma_f32_16x16x4_f32 ----
#pragma unroll
    for (int t = 0; t < 3; ++t) {
        const int j  = lane & 15;
        const int dd = t * 16 + j;
        const float dsv = 4.0f + (float)dd;
        const float p1x = bx + PR[2] * dsv;
        const float p1y = by + PR[5] * dsv;
        const float p1z = bz + PR[8] * dsv;
        // B column j = (x*z, y*z, z, 1); lanes 0-15 carry K=0/1, lanes 16-31 K=2/3
        v2f bop;
        if (lane < 16) { bop[0] = p1x * p1z; bop[1] = p1y * p1z; }
        else           { bop[0] = p1z;       bop[1] = 1.0f;      }

        v8f acc = {0.0f, 0.0f, 0.0f, 0.0f, 0.0f, 0.0f, 0.0f, 0.0f};
        acc = __builtin_amdgcn_wmma_f32_16x16x4_f32(
            /*neg_a=*/false, aop, /*neg_b=*/false, bop,
            /*c_mod=*/(short)0, acc, /*reuse_a=*/false, /*reuse_b=*/false);
        // D rows 0..2 (geom x,y,z) land in VGPRs 0..2 of lanes 0..15 (N = lane)
        if (lane < 16 && dd < DBINS) {
            float gxf = acc[0], gyf = acc[1], gzf = acc[2];
            int gx = (int)((gxf + 50.0f) * (1.0f / 3.0f));   // (geom - (BX - DX/2)) / DX
            int gy = (int)((gyf + 50.0f) * (1.0f / 3.0f));
            int gz = (int)((gzf + 5.0f)  * (1.0f / 3.0f));
            bool kept = (gx >= 0) & (gx < NXV) & (gy >= 0) & (gy < NYV) &
                        (gz >= 0) & (gz < NZV);
            // out flat index base: ((gz*C + c)*NXV + gx)*NYV + gy = base + c*PLANE
            unsigned segb = kept ? (unsigned)(gz * (CCH * PLANE) + gx * NYV + gy)
                                 : 0xFFFFFFFFu;
            sbuf[wv][dd] = segb;
        }
    }

    __syncthreads();

    // ---- run-length grouped scatter: consecutive d sharing a voxel flush once ----
    unsigned cur = 0xFFFFFFFFu;
    float ws = 0.0f;
    for (int dd = 0; dd < DBINS; ++dd) {
        unsigned s = sbuf[wv][dd];       // wave-uniform broadcast reads
        float wd = wbuf[wv][dd];
        if (s == cur) {
            ws += wd;
        } else {
            if (cur != 0xFFFFFFFFu) {
                atomicAdd(out + cur + (unsigned)lane * PLANE,        ws * ctx0);
                atomicAdd(out + cur + (unsigned)(lane + 32) * PLANE, ws * ctx1);
            }
            cur = s;
            ws = wd;
        }
    }
    if (cur != 0xFFFFFFFFu) {
        atomicAdd(out + cur + (unsigned)lane * PLANE,        ws * ctx0);
        atomicAdd(out + cur + (unsigned)(lane + 32) * PLANE, ws * ctx1);
    }
}

extern "C" void kernel_launch(void* const* d_in, const int* in_sizes, int n_in,
                              void* d_out, int out_size, void* d_ws, size_t ws_size,
                              hipStream_t stream) {
    const float* x          = (const float*)d_in[0];
    const float* rots       = (const float*)d_in[1];
    const float* trans      = (const float*)d_in[2];
    const float* intrins    = (const float*)d_in[3];
    const float* post_rots  = (const float*)d_in[4];
    const float* post_trans = (const float*)d_in[5];
    float* out  = (float*)d_out;
    float* cams = (float*)d_ws;   // 6 cameras * 24 floats

    lss_prep_kernel<<<1, 32, 0, stream>>>(rots, trans, intrins, post_rots, post_trans, cams);
    lss_zero_kernel<<<(out_size + 255) / 256, 256, 0, stream>>>(out, out_size);
    // 33600 pixels / 8 waves-per-block = 4200 blocks, 1 pixel per wave32
    lss_splat_kernel<<<NPIX / 8, 256, 0, stream>>>(x, cams, out);
}